// LightGCN_26491358281938
// MI455X (gfx1250) — compile-verified
//
#include <hip/hip_runtime.h>
#include <hip/hip_bf16.h>

#define N_USERS_C 100000
#define N_ITEMS_C 100000
#define NTOT_C    200000
#define DIM_C     64
#define NNZ_C     3200000
#define NPAIR_C   500000
#define RES_C     0.2f

typedef __attribute__((ext_vector_type(2))) float v2f;
typedef __attribute__((ext_vector_type(8))) float v8f;

// ---------------------------------------------------------------------------
// init: A = base, acc = base, Bb = 0.   (float4 per thread over N*D elems)
// base row i = user_table[i] for i < N_USERS else item_table[i - N_USERS]
// ---------------------------------------------------------------------------
__global__ void lg_init(const float* __restrict__ ut, const float* __restrict__ it,
                        float* __restrict__ A, float* __restrict__ Bb,
                        float* __restrict__ acc) {
    size_t i = (size_t)blockIdx.x * blockDim.x + threadIdx.x;   // float4 index
    const size_t total4 = (size_t)NTOT_C * DIM_C / 4;           // 3,200,000
    if (i >= total4) return;
    const size_t ubound4 = (size_t)N_USERS_C * DIM_C / 4;       // 1,600,000
    float4 b = (i < ubound4) ? ((const float4*)ut)[i]
                             : ((const float4*)it)[i - ubound4];
    ((float4*)A)[i]   = b;
    ((float4*)acc)[i] = b;
    ((float4*)Bb)[i]  = make_float4(0.f, 0.f, 0.f, 0.f);
}

// ---------------------------------------------------------------------------
// spmm: one wave per edge; 32 lanes x float2 = 64-dim row.
// y[rows[e]] += vals[e] * x[cols[e]]   (atomic f32 scatter into L2)
// ---------------------------------------------------------------------------
__global__ void lg_spmm(const int* __restrict__ rows, const int* __restrict__ cols,
                        const float* __restrict__ vals, const float* __restrict__ x,
                        float* __restrict__ y) {
    long long t = (long long)blockIdx.x * blockDim.x + threadIdx.x;
    int e    = (int)(t >> 5);
    int lane = (int)(t & 31);
    if (e >= NNZ_C) return;
    int   r = rows[e];
    int   c = cols[e];
    float v = vals[e];
    const float2 xv = ((const float2*)(x + (size_t)c * DIM_C))[lane];
    float* yp = y + (size_t)r * DIM_C + lane * 2;
    atomicAdd(yp,     v * xv.x);
    atomicAdd(yp + 1, v * xv.y);
}

// ---------------------------------------------------------------------------
// fuse: t = RES*base + (1-RES)*Bb ; A = t ; acc += t ; Bb = 0 (for next layer)
// ---------------------------------------------------------------------------
__global__ void lg_fuse(const float* __restrict__ ut, const float* __restrict__ it,
                        float* __restrict__ A, float* __restrict__ Bb,
                        float* __restrict__ acc) {
    size_t i = (size_t)blockIdx.x * blockDim.x + threadIdx.x;   // float4 index
    const size_t total4 = (size_t)NTOT_C * DIM_C / 4;
    if (i >= total4) return;
    const size_t ubound4 = (size_t)N_USERS_C * DIM_C / 4;
    float4 b = (i < ubound4) ? ((const float4*)ut)[i]
                             : ((const float4*)it)[i - ubound4];
    float4 s = ((const float4*)Bb)[i];
    float4 tv;
    tv.x = RES_C * b.x + (1.f - RES_C) * s.x;
    tv.y = RES_C * b.y + (1.f - RES_C) * s.y;
    tv.z = RES_C * b.z + (1.f - RES_C) * s.z;
    tv.w = RES_C * b.w + (1.f - RES_C) * s.w;
    ((float4*)A)[i] = tv;
    float4 a = ((const float4*)acc)[i];
    a.x += tv.x; a.y += tv.y; a.z += tv.z; a.w += tv.w;
    ((float4*)acc)[i] = a;
    ((float4*)Bb)[i]  = make_float4(0.f, 0.f, 0.f, 0.f);
}

// ---------------------------------------------------------------------------
// dot: per wave, 16 pairs.  out[p] = dot(acc[user[p]], acc[N_USERS+item[p]])/16
// via diag( U(16x64) * I^T(64x16) ) with 16x V_WMMA_F32_16X16X4_F32.
//
// f32 16x4 A fragment: lane half h = lane>>4, row M = lane&15,
//   (VGPR r, half h) -> K = 4k + 2h + r  => one float2 load at offset 4k+2h.
// f32 4x16 B fragment: col N = lane&15, same (r,h)->K striping => same load.
// 16x16 f32 D: lanes 0-15 hold (M=v, N=lane); lanes 16-31 hold (M=v+8, N=lane-16).
//   diag p<8  -> lane p,    VGPR p
//   diag p>=8 -> lane p+16, VGPR p-8   (lanes 24..31, VGPR lane-24)
// ---------------------------------------------------------------------------
__global__ void lg_dot(const float* __restrict__ acc, const int* __restrict__ users,
                       const int* __restrict__ items, float* __restrict__ out) {
    long long t = (long long)blockIdx.x * blockDim.x + threadIdx.x;
    int wave = (int)(t >> 5);
    int lane = (int)(threadIdx.x & 31);
    if (wave >= NPAIR_C / 16) return;        // 31250 exactly, wave-uniform guard
    int p = lane & 15;
    int h = lane >> 4;
    int pairBase = wave * 16;
    int urow = users[pairBase + p];
    int irow = N_USERS_C + items[pairBase + p];
    const float* up = acc + (size_t)urow * DIM_C;
    const float* ip = acc + (size_t)irow * DIM_C;

    v8f c = {};
#pragma unroll
    for (int k = 0; k < 16; ++k) {
        int off = k * 4 + h * 2;
        v2f a = *(const v2f*)(up + off);
        v2f b = *(const v2f*)(ip + off);
        // (neg_a, A, neg_b, B, c_mod, C, reuse_a, reuse_b)
        c = __builtin_amdgcn_wmma_f32_16x16x4_f32(
                false, a, false, b, (short)0, c, false, false);
    }

    const float scale = 0.0625f;  // (1/(K_LAYERS+1))^2 = (1/4)^2
    if (lane < 8) {
        out[pairBase + lane] = c[lane] * scale;
    } else if (lane >= 24) {
        out[pairBase + (lane - 16)] = c[lane - 24] * scale;
    }
}

// ---------------------------------------------------------------------------
extern "C" void kernel_launch(void* const* d_in, const int* in_sizes, int n_in,
                              void* d_out, int out_size, void* d_ws, size_t ws_size,
                              hipStream_t stream) {
    (void)in_sizes; (void)n_in; (void)out_size; (void)ws_size;
    const float* ut    = (const float*)d_in[0];   // user_table  (100000*64)
    const float* it    = (const float*)d_in[1];   // item_table  (100000*64)
    const int*   rows  = (const int*)  d_in[2];   // (3.2M)
    const int*   cols  = (const int*)  d_in[3];   // (3.2M)
    const float* vals  = (const float*)d_in[4];   // (3.2M)
    const int*   users = (const int*)  d_in[5];   // (500000)
    const int*   items = (const int*)  d_in[6];   // (500000)
    float*       out   = (float*)d_out;           // (500000)

    const size_t tableElems = (size_t)NTOT_C * DIM_C;   // 12.8M floats = 51.2 MB
    float* A   = (float*)d_ws;                // cur
    float* Bb  = A  + tableElems;             // spmm scatter target
    float* acc = Bb + tableElems;             // running sum (needs 153.6 MB ws)

    const int elem4Blocks = (int)(tableElems / 4 / 256);        // 12500
    const int spmmBlocks  = (int)(((size_t)NNZ_C * 32) / 256);  // 400000
    const int dotBlocks   = (NPAIR_C / 16 * 32 + 255) / 256;    // 3907

    lg_init<<<elem4Blocks, 256, 0, stream>>>(ut, it, A, Bb, acc);
    for (int layer = 0; layer < 3; ++layer) {
        lg_spmm<<<spmmBlocks, 256, 0, stream>>>(rows, cols, vals, A, Bb);
        lg_fuse<<<elem4Blocks, 256, 0, stream>>>(ut, it, A, Bb, acc);
    }
    lg_dot<<<dotBlocks, 256, 0, stream>>>(acc, users, items, out);
}